// GCN_80642305949972
// MI455X (gfx1250) — compile-verified
//
#include <hip/hip_runtime.h>
#include <math.h>

typedef float v2f __attribute__((ext_vector_type(2)));
typedef float v8f __attribute__((ext_vector_type(8)));

#define D 128
#define PSTR 288   // dwords per k-pair row of interleaved W: 256 + 32 pad (288%64==32 -> halves split banks)
#define XSTR 132   // dwords per X row: 128 + 4 pad (132%64==4 -> 16 lanes x step4 = all banks once)
#define GEMM_LDS_BYTES ((64 * PSTR + 128 * XSTR) * 4)   // 141312 B  (< 320KB/WGP)

// ---------------- degree / dinv ----------------
__global__ void k_fill1(float* __restrict__ deg, int n) {
    int i = blockIdx.x * blockDim.x + threadIdx.x;
    if (i < n) deg[i] = 1.0f;              // self-loop contributes 1 to in-degree
}

__global__ void k_deg(const long long* __restrict__ dst, float* __restrict__ deg, int e) {
    int i = blockIdx.x * blockDim.x + threadIdx.x;
    if (i < e) atomicAdd(&deg[(int)dst[i]], 1.0f);
}

__global__ void k_rsqrt_ip(float* __restrict__ d, int n) {
    int i = blockIdx.x * blockDim.x + threadIdx.x;
    if (i < n) d[i] = rsqrtf(fmaxf(d[i], 1.0f));
}

// ---------------- fp32 WMMA GEMM: H[n,128] = X[n,128] @ W[128,128] ----------------
// 256 threads = 8 waves; block computes a 128-row slab.
// W staged pair-interleaved so each B fragment is ONE aligned ds_load_b64.
// X staged via global_load_async_to_lds_b128 (ASYNCcnt path); edge rows are clamped
// (clamped rows produce garbage C rows that are never stored).
// K processed in groups of 8: 18 independent b64 LDS loads, then 16 WMMAs,
// so the dscnt wait amortizes over 16 matrix ops instead of 2.
__global__ __launch_bounds__(256) void k_gemm128(const float* __restrict__ X,
                                                 const float* __restrict__ W,
                                                 float* __restrict__ H, int n) {
    extern __shared__ float smem[];
    float* Wp = smem;                 // [64][PSTR]  pair-interleaved: Wp[k>>1][n*2 + (k&1)] = W[k][n]
    float* Xs = smem + 64 * PSTR;     // [128][XSTR] row-major

    const int tid  = threadIdx.x;
    const int row0 = blockIdx.x << 7;

    // ---- stage X slab: async memory->LDS (b128), clamped rows at grid edge ----
    for (int f = tid; f < 4096; f += 256) {
        int r = f >> 5, c = (f & 31) << 2;
        int gr = row0 + r;
        if (gr >= n) gr = n - 1;                       // clamp: result rows >= n are never stored
        const float* gp = X + (size_t)gr * D + c;
        unsigned lo = (unsigned)(unsigned long long)(Xs + r * XSTR + c);
        asm volatile("global_load_async_to_lds_b128 %0, %1, off"
                     :: "v"(lo), "v"(gp) : "memory");
    }

    // ---- stage W pair-interleaved (4 x ds_store_b32 per float4; once per block) ----
    for (int f = tid; f < 4096; f += 256) {
        int r = f >> 5, c = (f & 31) << 2;             // W[r][c..c+3]
        float4 w4 = *(const float4*)(W + r * D + c);
        float* dp = Wp + (r >> 1) * PSTR + (c << 1) + (r & 1);
        dp[0] = w4.x; dp[2] = w4.y; dp[4] = w4.z; dp[6] = w4.w;
    }

    asm volatile("s_wait_asynccnt 0x0" ::: "memory");
    __syncthreads();

    const int wave = tid >> 5;
    const int lane = tid & 31;
    const int l15  = lane & 15;
    const int half = lane >> 4;               // 0: K = k0,k0+1   1: K = k0+2,k0+3
    const int mrow = (wave << 4) + l15;       // A row (both half-waves carry M = l15)

    const float* aRow  = Xs + mrow * XSTR + (half << 1);   // + k0 each step
    const float* bBase = Wp + half * PSTR + (l15 << 1);    // + (k0>>1)*PSTR each step

    v8f acc[8] = {};

    for (int k0 = 0; k0 < D; k0 += 8) {
        // ---- issue all 18 b64 loads for this K-group into distinct registers ----
        v2f aA = *(const v2f*)(aRow + k0);
        v2f aB = *(const v2f*)(aRow + k0 + 4);
        const float* bpA = bBase + (k0 >> 1) * PSTR;
        const float* bpB = bpA + 2 * PSTR;
        v2f bA[8], bB[8];
#pragma unroll
        for (int t = 0; t < 8; ++t) bA[t] = *(const v2f*)(bpA + (t << 5));
#pragma unroll
        for (int t = 0; t < 8; ++t) bB[t] = *(const v2f*)(bpB + (t << 5));

        // ---- 16 back-to-back WMMAs ----
#pragma unroll
        for (int t = 0; t < 8; ++t)
            acc[t] = __builtin_amdgcn_wmma_f32_16x16x4_f32(
                false, aA, false, bA[t], (short)0, acc[t], false, false);
#pragma unroll
        for (int t = 0; t < 8; ++t)
            acc[t] = __builtin_amdgcn_wmma_f32_16x16x4_f32(
                false, aB, false, bB[t], (short)0, acc[t], false, false);
    }

    // C/D layout: VGPR r -> M = r + 8*half, N = l15.
#pragma unroll
    for (int t = 0; t < 8; ++t) {
#pragma unroll
        for (int r = 0; r < 8; ++r) {
            int gr = row0 + (wave << 4) + r + (half << 3);
            if (gr < n) H[(size_t)gr * D + (t << 4) + l15] = acc[t][r];
        }
    }
}

// ---------------- out = bias + self-loop term ----------------
__global__ void k_selfinit(const float* __restrict__ Hm, const float* __restrict__ dinv,
                           const float* __restrict__ bias, float* __restrict__ O, int n) {
    size_t idx = (size_t)blockIdx.x * blockDim.x + threadIdx.x;
    if (idx >= (size_t)n * D) return;
    int i = (int)(idx >> 7);
    float di = dinv[i];
    O[idx] = bias[idx & (D - 1)] + Hm[idx] * di * di;
}

// ---------------- edge aggregation: one wave32 per edge, 4 dims/lane ----------------
__global__ void k_edgeagg(const float* __restrict__ Hm, const float* __restrict__ dinv,
                          const long long* __restrict__ src, const long long* __restrict__ dst,
                          float* __restrict__ O, int e) {
    size_t gt = (size_t)blockIdx.x * blockDim.x + threadIdx.x;
    int eid = (int)(gt >> 5);
    if (eid >= e) return;
    int lane = threadIdx.x & 31;
    int s = (int)src[eid], d = (int)dst[eid];
    float norm = dinv[s] * dinv[d];
    float4 hv = *(const float4*)(Hm + (size_t)s * D + (lane << 2));   // coalesced 512B row (L2-resident)
    float* o = O + (size_t)d * D + (lane << 2);
    atomicAdd(o + 0, hv.x * norm);
    atomicAdd(o + 1, hv.y * norm);
    atomicAdd(o + 2, hv.z * norm);
    atomicAdd(o + 3, hv.w * norm);
}

__global__ void k_relu(float* __restrict__ x, size_t nd) {
    size_t idx = (size_t)blockIdx.x * blockDim.x + threadIdx.x;
    if (idx < nd) x[idx] = fmaxf(x[idx], 0.f);
}

// ---------------- layer 2: z[i] = dot(H[i,:], W2) ; wave per node ----------------
__global__ void k_dot(const float* __restrict__ Hm, const float* __restrict__ w,
                      float* __restrict__ z, int n) {
    size_t gt = (size_t)blockIdx.x * blockDim.x + threadIdx.x;
    int i = (int)(gt >> 5);
    if (i >= n) return;
    int lane = threadIdx.x & 31;
    float4 h4 = *(const float4*)(Hm + (size_t)i * D + (lane << 2));
    float4 w4 = *(const float4*)(w + (lane << 2));
    float p = h4.x * w4.x + h4.y * w4.y + h4.z * w4.z + h4.w * w4.w;
#pragma unroll
    for (int off = 16; off > 0; off >>= 1) p += __shfl_down(p, off, 32);
    if (lane == 0) z[i] = p;
}

__global__ void k_selfinit1(const float* __restrict__ z, const float* __restrict__ dinv,
                            const float* __restrict__ bias, float* __restrict__ O, int n) {
    int i = blockIdx.x * blockDim.x + threadIdx.x;
    if (i >= n) return;
    float di = dinv[i];
    O[i] = bias[0] + z[i] * di * di;
}

__global__ void k_edgescalar(const float* __restrict__ z, const float* __restrict__ dinv,
                             const long long* __restrict__ src, const long long* __restrict__ dst,
                             float* __restrict__ O, int e) {
    int i = blockIdx.x * blockDim.x + threadIdx.x;
    if (i >= e) return;
    int s = (int)src[i], d = (int)dst[i];
    atomicAdd(&O[d], z[s] * dinv[s] * dinv[d]);
}

__global__ void k_sigmoid(float* __restrict__ O, int n) {
    int i = blockIdx.x * blockDim.x + threadIdx.x;
    if (i < n) O[i] = 1.0f / (1.0f + expf(-O[i]));
}

// ---------------- driver ----------------
extern "C" void kernel_launch(void* const* d_in, const int* in_sizes, int n_in,
                              void* d_out, int out_size, void* d_ws, size_t ws_size,
                              hipStream_t stream) {
    const float*     x   = (const float*)d_in[0];
    const long long* ei  = (const long long*)d_in[1];   // int64 in reference
    const float*     W0  = (const float*)d_in[2];
    const float*     b0  = (const float*)d_in[3];
    const float*     W1  = (const float*)d_in[4];
    const float*     b1  = (const float*)d_in[5];
    const float*     W2  = (const float*)d_in[6];
    const float*     b2  = (const float*)d_in[7];
    float*           out = (float*)d_out;

    const int n = in_sizes[0] / D;
    const int e = in_sizes[1] / 2;
    const long long* src = ei;
    const long long* dst = ei + e;

    float* ws   = (float*)d_ws;
    size_t nAl  = ((size_t)n + 255) & ~(size_t)255;
    float* dinv = ws;                          // [n]
    float* bufA = ws + nAl;                    // [n*128]
    float* bufB = bufA + (size_t)n * D;        // [n*128]
    float* zbuf = bufA;                        // layer-2 scalar reuses bufA

    const int nBlk   = (n + 255) / 256;
    const int eBlk   = (e + 255) / 256;
    const int gBlk   = (n + 127) / 128;
    const size_t nd  = (size_t)n * D;
    const int ndBlk  = (int)((nd + 255) / 256);
    const int ewBlk  = (int)(((size_t)e * 32 + 255) / 256);   // wave per edge
    const int nwBlk  = (int)(((size_t)n * 32 + 255) / 256);   // wave per node

    // degrees -> dinv (in place)
    k_fill1   <<<nBlk, 256, 0, stream>>>(dinv, n);
    k_deg     <<<eBlk, 256, 0, stream>>>(dst, dinv, e);
    k_rsqrt_ip<<<nBlk, 256, 0, stream>>>(dinv, n);

    // layer 0
    k_gemm128 <<<gBlk, 256, GEMM_LDS_BYTES, stream>>>(x, W0, bufA, n);
    k_selfinit<<<ndBlk, 256, 0, stream>>>(bufA, dinv, b0, bufB, n);
    k_edgeagg <<<ewBlk, 256, 0, stream>>>(bufA, dinv, src, dst, bufB, e);
    k_relu    <<<ndBlk, 256, 0, stream>>>(bufB, nd);

    // layer 1
    k_gemm128 <<<gBlk, 256, GEMM_LDS_BYTES, stream>>>(bufB, W1, bufA, n);
    k_selfinit<<<ndBlk, 256, 0, stream>>>(bufA, dinv, b1, bufB, n);
    k_edgeagg <<<ewBlk, 256, 0, stream>>>(bufA, dinv, src, dst, bufB, e);
    k_relu    <<<ndBlk, 256, 0, stream>>>(bufB, nd);

    // layer 2 (D_OUT = 1)
    k_dot       <<<nwBlk, 256, 0, stream>>>(bufB, W2, zbuf, n);
    k_selfinit1 <<<nBlk, 256, 0, stream>>>(zbuf, dinv, b2, out, n);
    k_edgescalar<<<eBlk, 256, 0, stream>>>(zbuf, dinv, src, dst, out, e);
    k_sigmoid   <<<nBlk, 256, 0, stream>>>(out, n);
}